// YOLOv2Loss_48326972015289
// MI455X (gfx1250) — compile-verified
//
#include <hip/hip_runtime.h>
#include <math.h>

// Problem constants (from reference)
#define NBATCH 32
#define NANCH  5
#define NCLS   80
#define NG     16
#define HH     64
#define WW     64
#define HWC    4096          // H*W
#define CHAN   86            // 6 + NUM_CLASSES
#define CONF_TOT4 163840     // B*A*HW / 4 (float4 granularity)
#define NPAIR   (NBATCH*NG)  // 512
#define PI8     0.39269908169872414f // pi/8

// ws layout (floats), every slot below is fully overwritten each launch:
//   [0    .. 319 ] conf base-sum partials (320 = 5*64, WMMA-tile aligned)
//   [512  .. 1023] coord partials   (per pair)
//   [1024 .. 1535] conf-corr partials
//   [1536 .. 2047] cls NLL partials
//   [2048 .. 2559] theta partials
#define SEG_CONF  0
#define SEG_COORD 512
#define SEG_CORR  1024
#define SEG_CLS   1536
#define SEG_THETA 2048

__constant__ float c_aw[5] = {1.3221f, 3.19275f, 5.05587f, 9.47112f, 11.2364f};
__constant__ float c_ah[5] = {1.73145f, 4.00944f, 8.09892f, 4.84053f, 10.0071f};
__constant__ float c_aa[5] = {0.0f, 0.3927f, 0.7854f, 1.1781f, 1.5708f};

typedef float v2f __attribute__((ext_vector_type(2)));
typedef float v4f __attribute__((ext_vector_type(4)));
typedef float v8f __attribute__((ext_vector_type(8)));

// Fast sigmoid for the dense sweep: native v_exp_f32 + v_rcp_f32 (~1e-6 rel).
__device__ __forceinline__ float fast_sigmoid(float x) {
    float t = __expf(-x);
    return __builtin_amdgcn_rcpf(1.0f + t);
}
// Precise sigmoid for the 512 obj cells.
__device__ __forceinline__ float sigmoidf(float x) { return 1.0f / (1.0f + expf(-x)); }

// Full wave32 sum of 64 values (2 per lane) using V_WMMA_F32_16X16X4_F32 with
// B = ones: D[m,n] = rowsum(A[m,:]). Lane n holds rows 0..7 (lanes 0-15) or
// 8..15 (lanes 16-31) of column n in its 8 accumulators, so per-lane sum +
// xor-16 shuffle yields the grand total in every lane. Exact f32 FMA tree.
__device__ __forceinline__ float wave_wmma_sum(float a0, float a1) {
    v2f A; A.x = a0; A.y = a1;
    v2f Bv; Bv.x = 1.0f; Bv.y = 1.0f;
    v8f C = {};
    C = __builtin_amdgcn_wmma_f32_16x16x4_f32(false, A, false, Bv, (short)0, C, false, false);
    float s = C[0] + C[1] + C[2] + C[3] + C[4] + C[5] + C[6] + C[7];
    s += __shfl_xor(s, 16, 32);
    return s;
}

// ---------------------------------------------------------------------------
// Kernel A: dense sweep of the conf channel (the only term touching all
// cells). Base sum: sum over (b,a,hw) of (NOOBJ*sigmoid(conf))^2.
// float4 loads (rows are 4096 contiguous, 16B-aligned floats).
// 320 blocks x 256 threads x 2 float4 = 163840 vectors = 655360 elements.
// ---------------------------------------------------------------------------
__global__ void conf_sweep_kernel(const float* __restrict__ out,
                                  float* __restrict__ partial) {
    const int tid = blockIdx.x * blockDim.x + threadIdx.x;
    const int stride = gridDim.x * blockDim.x;   // 81920
    float acc = 0.0f;
    for (int e4 = tid; e4 < CONF_TOT4; e4 += stride) {
        int b   = e4 / (NANCH * HWC / 4);        // / 5120
        int r   = e4 - b * (NANCH * HWC / 4);
        int a   = r >> 10;
        int hw4 = r & 1023;
        size_t fidx = (size_t)(b * 430 + a * CHAN + 4) * HWC + (size_t)hw4 * 4;
        v4f x = *(const v4f*)(out + fidx);       // global_load_b128
        float s0 = fast_sigmoid(x.x);
        float s1 = fast_sigmoid(x.y);
        float s2 = fast_sigmoid(x.z);
        float s3 = fast_sigmoid(x.w);
        acc += s0 * s0 + s1 * s1 + s2 * s2 + s3 * s3;
    }
    // wave-level WMMA reduction, then 8 waves -> 1 via LDS
    __shared__ float lsum[8];
    const int lane = threadIdx.x & 31;
    const int wv   = threadIdx.x >> 5;
    float wtot = wave_wmma_sum(acc, 0.0f);
    if (lane == 0) lsum[wv] = wtot;
    __syncthreads();
    if (threadIdx.x == 0) {
        float t = 0.0f;
        for (int i = 0; i < 8; ++i) t += lsum[i];
        partial[SEG_CONF + blockIdx.x] = t;
    }
}

// ---------------------------------------------------------------------------
// Kernel B: one wave32 per ground-truth pair (b,g), 512 pairs.
// Recomputes build_targets for the pair (best anchor via cos argmax, cell,
// tx/ty/tw/th/tth), lane-parallel log-softmax over the 80 gathered logits,
// lane 0 gathers the 6 box channels. Writes 4 partial arrays into ws.
// ---------------------------------------------------------------------------
__global__ void obj_cells_kernel(const float* __restrict__ out,
                                 const float* __restrict__ gtb,
                                 const int* __restrict__ gtc,
                                 float* __restrict__ ws) {
    const int lane = threadIdx.x & 31;
    const int wv   = threadIdx.x >> 5;
    const int p    = blockIdx.x * 8 + wv;        // pair index 0..511
    const int b    = p >> 4;                     // p / G
    const float* gt = gtb + p * 5;

    const float gx  = gt[0] * (float)WW;
    const float gy  = gt[1] * (float)HH;
    const float gw  = gt[2] * (float)WW;
    const float gh  = gt[3] * (float)HH;
    const float gth = gt[4] * PI8;

    // argmax over anchors of cos(0.25*(gth - aa)); first-max on ties
    int best = 0;
    float bv = cosf(0.25f * (gth - c_aa[0]));
    #pragma unroll
    for (int a = 1; a < NANCH; ++a) {
        float v = cosf(0.25f * (gth - c_aa[a]));
        if (v > bv) { bv = v; best = a; }
    }

    int gi = (int)gx; gi = gi < 0 ? 0 : (gi > WW - 1 ? WW - 1 : gi);
    int gj = (int)gy; gj = gj < 0 ? 0 : (gj > HH - 1 ? HH - 1 : gj);
    const int cell = gj * WW + gi;
    const size_t base = (size_t)(b * 430 + best * CHAN) * HWC + cell;

    // ---- class NLL: logits are channels 6..85, stride HWC apart ----
    const float* cl = out + base + (size_t)6 * HWC;
    const float x0 = cl[(size_t)lane * HWC];
    const float x1 = cl[(size_t)(lane + 32) * HWC];
    const bool has2 = lane < 16;
    // clamp address so every lane issues a valid load; mask value after
    const float x2r = cl[(size_t)((has2 ? lane + 64 : 0)) * HWC];
    const float x2 = has2 ? x2r : -__builtin_inff();

    float m = fmaxf(x0, fmaxf(x1, x2));
    #pragma unroll
    for (int o = 16; o >= 1; o >>= 1) m = fmaxf(m, __shfl_xor(m, o, 32));

    float e = expf(x0 - m) + expf(x1 - m) + (has2 ? expf(x2 - m) : 0.0f);
    const int tc = gtc[p];
    float v = 0.0f;
    if (tc == lane)                   v = x0;
    else if (tc == lane + 32)         v = x1;
    else if (has2 && tc == lane + 64) v = x2;
    #pragma unroll
    for (int o = 16; o >= 1; o >>= 1) {
        e += __shfl_xor(e, o, 32);
        v += __shfl_xor(v, o, 32);
    }

    if (lane == 0) {
        const float nll = -(v - m - logf(e));

        const float ch0 = out[base];
        const float ch1 = out[base + (size_t)1 * HWC];
        const float ch2 = out[base + (size_t)2 * HWC];
        const float ch3 = out[base + (size_t)3 * HWC];
        const float ch4 = out[base + (size_t)4 * HWC];
        const float ch5 = out[base + (size_t)5 * HWC];

        const float tx  = gx - (float)gi;
        const float ty  = gy - (float)gj;
        const float tw  = logf(fmaxf(gw, 1.0f) / c_aw[best]);
        const float th_ = logf(fmaxf(gh, 1.0f) / c_ah[best]);
        const float tth = gth - c_aa[best];

        const float d0 = sigmoidf(ch0) - tx;
        const float d1 = sigmoidf(ch1) - ty;
        const float d2 = ch2 - tw;
        const float d3 = ch3 - th_;
        const float coordp = d0 * d0 + d1 * d1 + d2 * d2 + d3 * d3;

        // conf correction at obj cell: replace NOOBJ^2*s^2 with OBJ^2*(s-iou)^2
        const float s  = sigmoidf(ch4);
        const float dc = s - bv;                 // bv == iou_best
        const float confcorr = 25.0f * dc * dc - s * s;

        const float dth = fabsf(ch5 - tth);
        const float sl  = dth < 1.0f ? 0.5f * dth * dth : dth - 0.5f;

        ws[SEG_COORD + p] = coordp;
        ws[SEG_CORR  + p] = confcorr;
        ws[SEG_CLS   + p] = nll;
        ws[SEG_THETA + p] = sl;
    }
}

// ---------------------------------------------------------------------------
// Kernel C: one wave32. Reduce each partial segment with chained WMMA
// accumulations (64-value tiles into one f32 accumulator matrix), then
// combine into the 5 output scalars (reference return order).
// ---------------------------------------------------------------------------
__global__ void finalize_kernel(const float* __restrict__ ws,
                                float* __restrict__ outv) {
    const int lane = threadIdx.x;   // blockDim 32
    const int segoff[5]    = {SEG_CONF, SEG_COORD, SEG_CORR, SEG_CLS, SEG_THETA};
    const int segchunks[5] = {5, 8, 8, 8, 8};   // 320 or 512 values / 64
    float t[5];
    v2f Bv; Bv.x = 1.0f; Bv.y = 1.0f;
    #pragma unroll
    for (int s = 0; s < 5; ++s) {
        const float* seg = ws + segoff[s];
        v8f C = {};
        for (int i = 0; i < segchunks[s]; ++i) {
            v2f A;
            A.x = seg[i * 64 + lane];
            A.y = seg[i * 64 + 32 + lane];
            C = __builtin_amdgcn_wmma_f32_16x16x4_f32(false, A, false, Bv,
                                                      (short)0, C, false, false);
        }
        float sv = C[0] + C[1] + C[2] + C[3] + C[4] + C[5] + C[6] + C[7];
        sv += __shfl_xor(sv, 16, 32);
        t[s] = sv;
    }
    if (lane == 0) {
        const float loss_coord = 5.0f * t[1] / 2621440.0f;       // B*A*4*HW
        const float loss_conf  = (t[0] + t[2]) / 655360.0f;      // B*A*HW
        const float loss_cls   = 2.0f * t[3] / 512.0f;           // n = B*G
        const float loss_theta = 5.0f * t[4] / 512.0f;
        outv[0] = loss_coord + loss_conf + loss_cls + loss_theta;
        outv[1] = loss_coord;
        outv[2] = loss_conf;
        outv[3] = loss_cls;
        outv[4] = loss_theta;
    }
}

extern "C" void kernel_launch(void* const* d_in, const int* in_sizes, int n_in,
                              void* d_out, int out_size, void* d_ws, size_t ws_size,
                              hipStream_t stream) {
    const float* output = (const float*)d_in[0];   // (B, A*86, H, W) f32
    const float* gtb    = (const float*)d_in[1];   // (B, G, 5) f32
    const int*   gtc    = (const int*)d_in[2];     // (B, G) i32
    float* ws   = (float*)d_ws;                    // 2560 floats used (10 KB)
    float* outp = (float*)d_out;                   // 5 floats

    conf_sweep_kernel<<<320, 256, 0, stream>>>(output, ws);          // ws[0..319]
    obj_cells_kernel<<<64, 256, 0, stream>>>(output, gtb, gtc, ws);  // ws[512..2559]
    finalize_kernel<<<1, 32, 0, stream>>>(ws, outp);
}